// SEIRSDEModel_78855599555079
// MI455X (gfx1250) — compile-verified
//
#include <hip/hip_runtime.h>
#include <hip/hip_bf16.h>

// SEIR SDE Monte-Carlo: 1 thread = 1 trajectory, 1000 sequential Euler-Maruyama
// steps, streaming [nsteps, 4, B] f32 output (500 MiB) with non-temporal stores.
// dW[nsteps,4] is shared by all trajectories -> staged once per workgroup into
// LDS by the CDNA5 Tensor Data Mover (tensor_load_to_lds + s_wait_tensorcnt).

#define NPOP_F    10000.0f
#define DT_F      0.01f
#define EPS_F     1e-6f
#define MAX_STEPS 1000
#define BLOCK     256

typedef unsigned int u32x4 __attribute__((ext_vector_type(4)));
typedef int          i32x4 __attribute__((ext_vector_type(4)));
typedef int          i32x8 __attribute__((ext_vector_type(8)));

__global__ __launch_bounds__(BLOCK) void seir_sde_kernel(
    const float* __restrict__ beta,  const float* __restrict__ sigma,
    const float* __restrict__ gamma,
    const float* __restrict__ S0,    const float* __restrict__ E0,
    const float* __restrict__ I0,    const float* __restrict__ R0,
    const float* __restrict__ dW,
    float* __restrict__ out, int B, int nsteps)
{
    __shared__ float4 s_dw[MAX_STEPS];   // 16000 B of 320 KB WGP LDS

    const unsigned tid = threadIdx.x;
    const unsigned n4  = (unsigned)nsteps * 4u;   // total floats in dW table

    // ---- Stage dW into LDS with the Tensor Data Mover (wave 0 only). ----
    // D# per CDNA5 ISA: group0 = {count, lds_addr,
    // global_addr, type=2}; group1 = {data_size=4B, tensor_dim0=n4,
    // tensor_dim1=1, tile_dim0=n4, tile_dim1=1, stride0=n4}. Not in a
    // cluster -> workgroup_mask = 0. Groups 2/3 unused (<=2D tensor).
    if (tid < 32u) {
        const unsigned long long ga = (unsigned long long)(uintptr_t)dW;
        const unsigned lds_off = (unsigned)(uintptr_t)(&s_dw[0]); // low 32b of
                                                                  // generic LDS ptr
        u32x4 g0;
        g0[0] = 1u;                                    // count=1, user mode
        g0[1] = lds_off;                               // lds_addr (bytes)
        g0[2] = (unsigned)ga;                          // global_addr[31:0]
        g0[3] = (unsigned)((ga >> 32) & 0x01FFFFFFull) // global_addr[56:32]
              | (2u << 30);                            // type = 2 ("image")
        i32x8 g1;
        g1[0] = (int)(2u << 16);                       // data_size = 2 -> 4 bytes
        g1[1] = (int)((n4 & 0xFFFFu) << 16);           // tensor_dim0[15:0]
        g1[2] = (int)((n4 >> 16) | (1u << 16));        // tensor_dim0[31:16] | tensor_dim1=1
        g1[3] = (int)((n4 & 0xFFFFu) << 16);           // tile_dim0 = n4
        g1[4] = 1;                                     // tile_dim1 = 1, tile_dim2 = 0
        g1[5] = (int)n4;                               // tensor_dim0_stride[31:0]
        g1[6] = (int)((n4 & 0xFFFFu) << 16);           // stride0 hi=0 | stride1[15:0]
        g1[7] = (int)(n4 >> 16);                       // stride1[47:16]
        const i32x4 gz4 = {0, 0, 0, 0};
        const i32x8 gz8 = {0, 0, 0, 0, 0, 0, 0, 0};
        // amdgpu-toolchain (clang-23) 6-arg form:
        // (u32x4 g0, i32x8 g1, i32x4, i32x4, i32x8, i32 cpol)
        __builtin_amdgcn_tensor_load_to_lds(g0, g1, gz4, gz4, gz8, 0);
        __builtin_amdgcn_s_wait_tensorcnt(0);          // TENSORcnt == 0
    }
    __syncthreads();

    // One-time pre-scale by sqrt(dt) (bit-identical to reference's dWs table).
    const float sqdt = sqrtf(DT_F);
    for (unsigned i = tid; i < (unsigned)nsteps; i += BLOCK) {
        float4 w = s_dw[i];
        w.x *= sqdt; w.y *= sqdt; w.z *= sqdt; w.w *= sqdt;
        s_dw[i] = w;
    }
    __syncthreads();

    const unsigned g = blockIdx.x * BLOCK + tid;
    if (g >= (unsigned)B) return;   // no barriers past this point

    const float b  = beta[0];
    const float sg = sigma[0];
    const float gm = gamma[0];

    float S = S0[g], E = E0[g], I = I0[g], R = R0[g];

    const size_t Bs = (size_t)B;
    float* __restrict__ o0 = out + g;

    for (int t = 0; t < nsteps; ++t) {
        // clamp (reference: maximum(x, 1e-6) at step entry)
        S = fmaxf(S, EPS_F); E = fmaxf(E, EPS_F);
        I = fmaxf(I, EPS_F); R = fmaxf(R, EPS_F);

        const float inf_rate = b * S * I / NPOP_F;   // beta*S*I/N
        const float exp_rate = sg * E;               // sigma*E
        const float rec_rate = gm * I;               // gamma*I

        const float n_inf = sqrtf(inf_rate);
        const float n_exp = sqrtf(exp_rate);
        const float n_rec = sqrtf(rec_rate);

        const float4 w = s_dw[t];                    // ds_load_b128 broadcast

        const float Sn = S + (-inf_rate * DT_F) - n_inf * w.x;
        const float En = E + (inf_rate - exp_rate) * DT_F + n_inf * w.x - n_exp * w.y;
        const float In = I + (exp_rate - rec_rate) * DT_F + n_exp * w.y - n_rec * w.z;
        const float Rn = R + rec_rate * DT_F + n_rec * w.z;
        S = Sn; E = En; I = In; R = Rn;

        // out[t][state][g] — coalesced, streamed, never re-read -> NT stores
        float* p = o0 + (size_t)t * (Bs * 4u);
        __builtin_nontemporal_store(S, p);
        __builtin_nontemporal_store(E, p + Bs);
        __builtin_nontemporal_store(I, p + 2u * Bs);
        __builtin_nontemporal_store(R, p + 3u * Bs);
    }
}

extern "C" void kernel_launch(void* const* d_in, const int* in_sizes, int n_in,
                              void* d_out, int out_size, void* d_ws, size_t ws_size,
                              hipStream_t stream) {
    (void)n_in; (void)d_ws; (void)ws_size;
    const float* beta  = (const float*)d_in[0];
    const float* sigma = (const float*)d_in[1];
    const float* gamma = (const float*)d_in[2];
    const float* S0    = (const float*)d_in[3];
    const float* E0    = (const float*)d_in[4];
    const float* I0    = (const float*)d_in[5];
    const float* R0    = (const float*)d_in[6];
    const float* dW    = (const float*)d_in[7];
    float* out = (float*)d_out;

    const int B      = in_sizes[3];          // 32768 trajectories
    const int nsteps = in_sizes[7] / 4;      // 1000 steps
    (void)out_size;

    const dim3 grid((B + BLOCK - 1) / BLOCK);
    seir_sde_kernel<<<grid, BLOCK, 0, stream>>>(beta, sigma, gamma,
                                                S0, E0, I0, R0, dW,
                                                out, B, nsteps);
}